// SelfAttentionPooling_43258910605917
// MI455X (gfx1250) — compile-verified
//
#include <hip/hip_runtime.h>
#include <hip/hip_bf16.h>

typedef __attribute__((ext_vector_type(2))) float v2f;
typedef __attribute__((ext_vector_type(8))) float v8f;

#define G_    1024
#define NPER_ 128
#define D_    256
#define KSEL_ 64

#if __has_builtin(__builtin_amdgcn_wmma_f32_16x16x4_f32)
#define HAVE_WMMA_F32X4 1
#endif

// LDS layout (dynamic shared):
//   [0,16384)        adjacency as u8 (128x128)
//   +16384: float sW[256], sSup[128], sScore[128], sSel[64]; int sIdx[64]
__global__ __launch_bounds__(256)
void sagpool_fused(const float* __restrict__ X,
                   const float* __restrict__ Adj,
                   const int* __restrict__ GB,
                   const float* __restrict__ W,
                   const float* __restrict__ Bias,
                   float* __restrict__ out1,
                   float* __restrict__ out2,
                   float* __restrict__ out3)
{
    extern __shared__ unsigned char smem[];
    unsigned char* sAdj  = smem;                       // 16384 B
    float* sW     = (float*)(smem + 16384);            // 256
    float* sSup   = sW + 256;                          // 128
    float* sScore = sSup + 128;                        // 128
    float* sSel   = sScore + 128;                      // 64
    int*   sIdx   = (int*)(sSel + 64);                 // 64

    const int g    = blockIdx.x;
    const int t    = threadIdx.x;
    const int lane = t & 31;
    const int wv   = t >> 5;

    // ---- W -> LDS (256 floats, one per thread) ----
    sW[t] = W[t];

    // ---- adjacency -> LDS as bytes (exactly 0/1 in input) ----
    {
        const float4* A4 = (const float4*)(Adj + (size_t)g * NPER_ * NPER_);
        for (int it = 0; it < 16; ++it) {
            int e4 = it * 256 + t;            // float4 index in [0,4096)
            float4 a = A4[e4];
            int base = e4 * 4;
            sAdj[base + 0] = (unsigned char)a.x;
            sAdj[base + 1] = (unsigned char)a.y;
            sAdj[base + 2] = (unsigned char)a.z;
            sAdj[base + 3] = (unsigned char)a.w;
        }
    }
    __syncthreads();

    // ---- support = X @ W + b  (8 waves x 16 rows, float4 coalesced) ----
    {
        const float bias = Bias[0];
        const float4* sW4 = (const float4*)sW;   // 64 x float4
        for (int i = 0; i < 16; ++i) {
            int n = wv * 16 + i;
            const float4* Xr = (const float4*)(X + ((size_t)g * NPER_ + n) * D_);
            float4 x0 = Xr[lane];
            float4 x1 = Xr[32 + lane];
            float4 w0 = sW4[lane];
            float4 w1 = sW4[32 + lane];
            float acc = x0.x*w0.x + x0.y*w0.y + x0.z*w0.z + x0.w*w0.w
                      + x1.x*w1.x + x1.y*w1.y + x1.z*w1.z + x1.w*w1.w;
            #pragma unroll
            for (int off = 16; off > 0; off >>= 1)
                acc += __shfl_down(acc, off, 32);
            if (lane == 0) sSup[n] = acc + bias;
        }
    }
    __syncthreads();

    // ---- node_score = tanh(A . support) ----
#ifdef HAVE_WMMA_F32X4
    {
        // Wave wv computes rows m0..m0+15 via V_WMMA_F32_16X16X4_F32.
        // A tile 16x4 (f32): lanes 0-15 hold M=0..15 with K={k,k+1},
        // lanes 16-31 hold K={k+2,k+3}. B 4x16 has all 16 columns equal
        // to the support chunk, so every column of D equals A.s.
        const int m0  = wv * 16;
        const int row = m0 + (lane & 15);
        const int kh  = (lane >> 4) * 2;          // 0 or 2
        const unsigned char* Ar = sAdj + row * NPER_;
        v8f c = {};
        for (int kb = 0; kb < NPER_; kb += 4) {
            int k0 = kb + kh;
            v2f a;  a.x  = (float)Ar[k0];  a.y  = (float)Ar[k0 + 1];
            v2f bv; bv.x = sSup[k0];       bv.y = sSup[k0 + 1];
            c = __builtin_amdgcn_wmma_f32_16x16x4_f32(
                    false, a, false, bv, (short)0, c, false, false);
        }
        // D layout: VGPR v, lanes 0-15 -> M=m0+v, lanes 16-31 -> M=m0+8+v.
        // Column 0 lives in lanes 0 and 16.
        if ((lane & 15) == 0) {
            int rbase = m0 + (lane >> 4) * 8;
            #pragma unroll
            for (int v = 0; v < 8; ++v)
                sScore[rbase + v] = tanhf(c[v]);
        }
    }
#else
    if (t < NPER_) {
        const unsigned char* Ar = sAdj + t * NPER_;
        float acc = 0.0f;
        for (int j = 0; j < NPER_; ++j)
            acc += (float)Ar[j] * sSup[j];
        sScore[t] = tanhf(acc);
    }
#endif
    __syncthreads();

    // ---- stable top-K (matches jax.lax.top_k: desc value, tie -> lower idx) ----
    if (t < NPER_) {
        float my = sScore[t];
        int rank = 0;
        for (int j = 0; j < NPER_; ++j) {
            float sj = sScore[j];
            rank += (sj > my || (sj == my && j < t)) ? 1 : 0;
        }
        if (rank < KSEL_) { sIdx[rank] = t; sSel[rank] = my; }
    }
    __syncthreads();

    // ---- mask_X = X[idx] * score  (float4 coalesced; 4 rows per pass) ----
    {
        const float4* X4 = (const float4*)X;
        float4* O4 = (float4*)out1;
        int sub = t >> 6;        // 0..3
        int col = t & 63;        // float4 column within a 256-float row
        for (int it = 0; it < 16; ++it) {
            int r = it * 4 + sub;
            int n = sIdx[r];
            float sc = sSel[r];
            float4 x = X4[((size_t)g * NPER_ + n) * (D_ / 4) + col];
            x.x *= sc; x.y *= sc; x.z *= sc; x.w *= sc;
            O4[((size_t)g * KSEL_ + r) * (D_ / 4) + col] = x;
        }
    }

    // ---- mask_adjacency = A[idx_i, idx_j]  (from LDS bytes) ----
    {
        float* o2 = out2 + (size_t)g * KSEL_ * KSEL_;
        for (int it = 0; it < 16; ++it) {
            int e = it * 256 + t;        // [0,4096)
            int i = e >> 6, j = e & 63;
            o2[e] = (float)sAdj[sIdx[i] * NPER_ + sIdx[j]];
        }
    }

    // ---- mask_graph_batch ----
    if (t < KSEL_)
        out3[(size_t)g * KSEL_ + t] = (float)GB[(size_t)g * NPER_ + sIdx[t]];
}

extern "C" void kernel_launch(void* const* d_in, const int* in_sizes, int n_in,
                              void* d_out, int out_size, void* d_ws, size_t ws_size,
                              hipStream_t stream) {
    const float* X   = (const float*)d_in[0];
    const float* Adj = (const float*)d_in[1];
    const int*   GB  = (const int*)d_in[2];
    const float* W   = (const float*)d_in[3];
    const float* B   = (const float*)d_in[4];

    float* out1 = (float*)d_out;                                  // (G*K, D)
    float* out2 = out1 + (size_t)G_ * KSEL_ * D_;                 // (G, K, K)
    float* out3 = out2 + (size_t)G_ * KSEL_ * KSEL_;              // (G*K,)

    size_t shmem = 16384 + (256 + 128 + 128 + 64) * sizeof(float)
                 + 64 * sizeof(int);                              // 18944 B

    sagpool_fused<<<G_, 256, shmem, stream>>>(X, Adj, GB, W, B, out1, out2, out3);
}